// ICP_50070728737433
// MI455X (gfx1250) — compile-verified
//
#include <hip/hip_runtime.h>
#include <math.h>

// ---------------------------------------------------------------------------
// ICP on MI455X (gfx1250): NN via V_WMMA_F32_16X16X4_F32 (K=4, z-padded 3D),
// Kabsch via Newton polar iteration, fixed 26-iteration launch sequence.
// ---------------------------------------------------------------------------

typedef float v2f __attribute__((ext_vector_type(2)));
typedef float v8f __attribute__((ext_vector_type(8)));

#define NPTS   8192
#define NITERS 26            // MAX_ITERS + 1
#define TOL    1e-4f
#define CHUNK  2048
#define TILES_PER_CHUNK (CHUNK / 16)
#define ROWS_PER_BLOCK  128  // 8 waves * 16 rows
#define NN_BLOCKS (NPTS / ROWS_PER_BLOCK)

// workspace layout (floats):
//   ws[0..15]   : accumulators  [0]=sumDist [1..3]=sum(pc) [4..6]=sum(p2sel)
//                 [7..15]=sum( sel_a * pc_b ) row-major
//   ws[16..18]  : state: prev_err, done, apply
//   ws[32..40]  : R (row-major)
//   ws[41..43]  : t
//   ws[48.. )   : pc array (8192*3 floats)
#define WS_ACC   0
#define WS_STATE 16
#define WS_R     32
#define WS_T     41
#define WS_PC    48

// ---------------------------------------------------------------------------
// init: pc = p1, zero accumulators + state
// ---------------------------------------------------------------------------
__global__ void icp_init(const float* __restrict__ p1, float* __restrict__ ws) {
    int i = blockIdx.x * blockDim.x + threadIdx.x;
    if (i < NPTS * 3) ws[WS_PC + i] = p1[i];
    if (blockIdx.x == 0 && threadIdx.x < 32) {
        if (threadIdx.x < 16) ws[WS_ACC + threadIdx.x] = 0.0f;
        if (threadIdx.x == 16) { ws[WS_STATE + 0] = 0.0f; }  // prev_err
        if (threadIdx.x == 17) { ws[WS_STATE + 1] = 0.0f; }  // done
        if (threadIdx.x == 18) { ws[WS_STATE + 2] = 1.0f; }  // apply
    }
}

// ---------------------------------------------------------------------------
// NN kernel: per-row argmin over 8192 p2 points using f32 WMMA 16x16x4.
// Accumulates the 16 reduction terms into ws accumulators.
// ---------------------------------------------------------------------------
__global__ __launch_bounds__(256) void icp_nn(const float* __restrict__ ws_pc,
                                              const float* __restrict__ p2,
                                              float* __restrict__ acc) {
    __shared__ float4 sP[CHUNK];          // x, y, z, 0.5*||p||^2   (32 KB)
    __shared__ float  redV[8][16][16];    // ( 8 KB)
    __shared__ int    redI[8][16][16];    // ( 8 KB)
    __shared__ float  blkAcc[16];

    const int tid  = threadIdx.x;
    const int wave = tid >> 5;
    const int lane = tid & 31;
    const int slot = lane & 15;
    const int half = lane >> 4;
    const int blockBase = blockIdx.x * ROWS_PER_BLOCK;
    const int rowBase   = blockBase + wave * 16;
    const int myRow     = rowBase + slot;           // A-matrix row for this lane

    // A operand (16x4 f32): half0 -> (x,y) for K=0,1 ; half1 -> (z,0) for K=2,3
    float ax = ws_pc[myRow * 3 + 0];
    float ay = ws_pc[myRow * 3 + 1];
    float az = ws_pc[myRow * 3 + 2];
    v2f a;
    a.x = half ? az : ax;
    a.y = half ? 0.0f : ay;

    // track max of m = S - 0.5*||b||^2  (equivalent to min of d2, na constant)
    float bestV[8];
    int   bestI[8];
#pragma unroll
    for (int r = 0; r < 8; ++r) { bestV[r] = -3.0e38f; bestI[r] = 0; }

    for (int chunk = 0; chunk < NPTS; chunk += CHUNK) {
        // cooperative stage of p2 chunk into LDS
        for (int i = tid; i < CHUNK; i += 256) {
            float x = p2[(chunk + i) * 3 + 0];
            float y = p2[(chunk + i) * 3 + 1];
            float z = p2[(chunk + i) * 3 + 2];
            sP[i] = make_float4(x, y, z, 0.5f * (x * x + y * y + z * z));
        }
        __syncthreads();

        for (int t = 0; t < TILES_PER_CHUNK; ++t) {
            float4 q = sP[t * 16 + slot];
            v2f b;
            b.x = half ? q.z : q.x;
            b.y = half ? 0.0f : q.y;
            v8f c = {};
            // D = A(16x4) x B(4x16) + 0   -> v_wmma_f32_16x16x4_f32
            c = __builtin_amdgcn_wmma_f32_16x16x4_f32(
                    false, a, false, b, (short)0, c, false, false);
            int j = chunk + t * 16 + slot;
#pragma unroll
            for (int r = 0; r < 8; ++r) {
                float m = c[r] - q.w;
                bool better = (m > bestV[r]);
                bestV[r] = better ? m : bestV[r];
                bestI[r] = better ? j : bestI[r];
            }
        }
        __syncthreads();
    }

    // element (r, lane) is row (r + 8*half) of this wave's tile, column slot
#pragma unroll
    for (int r = 0; r < 8; ++r) {
        int mrel = r + 8 * half;
        redV[wave][mrel][slot] = bestV[r];
        redI[wave][mrel][slot] = bestI[r];
    }
    if (tid < 16) blkAcc[tid] = 0.0f;
    __syncthreads();

    if (tid < ROWS_PER_BLOCK) {
        int w = tid >> 4, mrel = tid & 15;
        float bV = redV[w][mrel][0];
        int   bI = redI[w][mrel][0];
        for (int s = 1; s < 16; ++s) {
            float v = redV[w][mrel][s];
            int   i2 = redI[w][mrel][s];
            if (v > bV || (v == bV && i2 < bI)) { bV = v; bI = i2; }
        }
        int grow = blockBase + w * 16 + mrel;
        float px = ws_pc[grow * 3 + 0];
        float py = ws_pc[grow * 3 + 1];
        float pz = ws_pc[grow * 3 + 2];
        float na = px * px + py * py + pz * pz;
        float d2 = na - 2.0f * bV;
        float dist = sqrtf(fmaxf(d2, 0.0f));
        float sx = p2[bI * 3 + 0];
        float sy = p2[bI * 3 + 1];
        float sz = p2[bI * 3 + 2];
        atomicAdd(&blkAcc[0], dist);
        atomicAdd(&blkAcc[1], px); atomicAdd(&blkAcc[2], py); atomicAdd(&blkAcc[3], pz);
        atomicAdd(&blkAcc[4], sx); atomicAdd(&blkAcc[5], sy); atomicAdd(&blkAcc[6], sz);
        atomicAdd(&blkAcc[7 + 0], sx * px); atomicAdd(&blkAcc[7 + 1], sx * py); atomicAdd(&blkAcc[7 + 2], sx * pz);
        atomicAdd(&blkAcc[7 + 3], sy * px); atomicAdd(&blkAcc[7 + 4], sy * py); atomicAdd(&blkAcc[7 + 5], sy * pz);
        atomicAdd(&blkAcc[7 + 6], sz * px); atomicAdd(&blkAcc[7 + 7], sz * py); atomicAdd(&blkAcc[7 + 8], sz * pz);
    }
    __syncthreads();
    if (tid < 16) atomicAdd(&acc[tid], blkAcc[tid]);
}

// ---------------------------------------------------------------------------
// polar factor of 3x3 (== U V^T of SVD) via scaled Newton iteration
// ---------------------------------------------------------------------------
__device__ inline void polar3(const float* H, float* R) {
    float X[9];
#pragma unroll
    for (int k = 0; k < 9; ++k) X[k] = H[k];
    for (int it = 0; it < 30; ++it) {
        float c00 =  (X[4] * X[8] - X[5] * X[7]);
        float c01 = -(X[3] * X[8] - X[5] * X[6]);
        float c02 =  (X[3] * X[7] - X[4] * X[6]);
        float c10 = -(X[1] * X[8] - X[2] * X[7]);
        float c11 =  (X[0] * X[8] - X[2] * X[6]);
        float c12 = -(X[0] * X[7] - X[1] * X[6]);
        float c20 =  (X[1] * X[5] - X[2] * X[4]);
        float c21 = -(X[0] * X[5] - X[2] * X[3]);
        float c22 =  (X[0] * X[4] - X[1] * X[3]);
        float det = X[0] * c00 + X[1] * c01 + X[2] * c02;
        if (fabsf(det) < 1e-30f) det = (det < 0.0f) ? -1e-30f : 1e-30f;
        float invT[9] = { c00 / det, c01 / det, c02 / det,
                          c10 / det, c11 / det, c12 / det,
                          c20 / det, c21 / det, c22 / det };  // = X^{-T}
        float fX = 0.0f, fI = 0.0f;
#pragma unroll
        for (int k = 0; k < 9; ++k) { fX += X[k] * X[k]; fI += invT[k] * invT[k]; }
        float mu = sqrtf(sqrtf(fI / fX));  // (||X^-1||_F / ||X||_F)^(1/2)
#pragma unroll
        for (int k = 0; k < 9; ++k) X[k] = 0.5f * (mu * X[k] + invT[k] / mu);
    }
#pragma unroll
    for (int k = 0; k < 9; ++k) R[k] = X[k];
}

// ---------------------------------------------------------------------------
// per-iteration solve (1 thread): Kabsch from accumulators + done logic
// ---------------------------------------------------------------------------
__global__ void icp_solve(float* __restrict__ ws) {
    if (threadIdx.x != 0 || blockIdx.x != 0) return;
    float* acc = ws + WS_ACC;
    const float N = (float)NPTS;
    float mean_err = acc[0] / N;
    float c1[3], c2[3];
#pragma unroll
    for (int a2 = 0; a2 < 3; ++a2) { c1[a2] = acc[1 + a2] / N; c2[a2] = acc[4 + a2] / N; }
    float H[9];
#pragma unroll
    for (int a2 = 0; a2 < 3; ++a2)
#pragma unroll
        for (int b2 = 0; b2 < 3; ++b2)
            H[a2 * 3 + b2] = acc[7 + a2 * 3 + b2] - N * c2[a2] * c1[b2];
    float Rm[9];
    polar3(H, Rm);
    float tv[3];
#pragma unroll
    for (int a2 = 0; a2 < 3; ++a2)
        tv[a2] = c2[a2] - (Rm[a2 * 3 + 0] * c1[0] + Rm[a2 * 3 + 1] * c1[1] + Rm[a2 * 3 + 2] * c1[2]);

    float prev = ws[WS_STATE + 0];
    float done = ws[WS_STATE + 1];
    float apply = (done == 0.0f) ? 1.0f : 0.0f;
    float newdone = (done != 0.0f || fabsf(prev - mean_err) < TOL) ? 1.0f : 0.0f;
    ws[WS_STATE + 0] = (done != 0.0f) ? prev : mean_err;
    ws[WS_STATE + 1] = newdone;
    ws[WS_STATE + 2] = apply;
#pragma unroll
    for (int k = 0; k < 9; ++k) ws[WS_R + k] = Rm[k];
#pragma unroll
    for (int k = 0; k < 3; ++k) ws[WS_T + k] = tv[k];
#pragma unroll
    for (int k = 0; k < 16; ++k) acc[k] = 0.0f;  // ready for next reduction
}

// ---------------------------------------------------------------------------
// apply: pc = apply ? R*pc + t : pc
// ---------------------------------------------------------------------------
__global__ void icp_apply(float* __restrict__ ws) {
    float apply = ws[WS_STATE + 2];
    int i = blockIdx.x * blockDim.x + threadIdx.x;
    if (i < NPTS && apply != 0.0f) {
        float* pc = ws + WS_PC;
        float x = pc[i * 3 + 0], y = pc[i * 3 + 1], z = pc[i * 3 + 2];
        float nx = ws[WS_R + 0] * x + ws[WS_R + 1] * y + ws[WS_R + 2] * z + ws[WS_T + 0];
        float ny = ws[WS_R + 3] * x + ws[WS_R + 4] * y + ws[WS_R + 5] * z + ws[WS_T + 1];
        float nz = ws[WS_R + 6] * x + ws[WS_R + 7] * y + ws[WS_R + 8] * z + ws[WS_T + 2];
        pc[i * 3 + 0] = nx; pc[i * 3 + 1] = ny; pc[i * 3 + 2] = nz;
    }
}

// ---------------------------------------------------------------------------
// final reduction for _get_transform(p1, pc): Σp1, Σpc, Σ pc_a * p1_b
// ---------------------------------------------------------------------------
__global__ __launch_bounds__(256) void icp_final_reduce(const float* __restrict__ p1,
                                                        float* __restrict__ ws) {
    __shared__ float blkAcc[15];
    int tid = threadIdx.x;
    if (tid < 15) blkAcc[tid] = 0.0f;
    __syncthreads();
    int i = blockIdx.x * blockDim.x + tid;
    if (i < NPTS) {
        const float* pc = ws + WS_PC;
        float ax = p1[i * 3 + 0], ay = p1[i * 3 + 1], az = p1[i * 3 + 2];
        float bx = pc[i * 3 + 0], by = pc[i * 3 + 1], bz = pc[i * 3 + 2];
        atomicAdd(&blkAcc[0], ax); atomicAdd(&blkAcc[1], ay); atomicAdd(&blkAcc[2], az);
        atomicAdd(&blkAcc[3], bx); atomicAdd(&blkAcc[4], by); atomicAdd(&blkAcc[5], bz);
        atomicAdd(&blkAcc[6 + 0], bx * ax); atomicAdd(&blkAcc[6 + 1], bx * ay); atomicAdd(&blkAcc[6 + 2], bx * az);
        atomicAdd(&blkAcc[6 + 3], by * ax); atomicAdd(&blkAcc[6 + 4], by * ay); atomicAdd(&blkAcc[6 + 5], by * az);
        atomicAdd(&blkAcc[6 + 6], bz * ax); atomicAdd(&blkAcc[6 + 7], bz * ay); atomicAdd(&blkAcc[6 + 8], bz * az);
    }
    __syncthreads();
    if (tid < 15) atomicAdd(&ws[WS_ACC + 1 + tid], blkAcc[tid]);
}

// ---------------------------------------------------------------------------
// final solve (1 thread): R,t from (p1, pc); emit 4x4 T
// ---------------------------------------------------------------------------
__global__ void icp_final_solve(const float* __restrict__ ws, float* __restrict__ out) {
    if (threadIdx.x != 0 || blockIdx.x != 0) return;
    const float* acc = ws + WS_ACC;
    const float N = (float)NPTS;
    float c1[3], c2[3];
#pragma unroll
    for (int a2 = 0; a2 < 3; ++a2) { c1[a2] = acc[1 + a2] / N; c2[a2] = acc[4 + a2] / N; }
    float H[9];
#pragma unroll
    for (int a2 = 0; a2 < 3; ++a2)
#pragma unroll
        for (int b2 = 0; b2 < 3; ++b2)
            H[a2 * 3 + b2] = acc[7 + a2 * 3 + b2] - N * c2[a2] * c1[b2];
    float Rm[9];
    polar3(H, Rm);
    float tv[3];
#pragma unroll
    for (int a2 = 0; a2 < 3; ++a2)
        tv[a2] = c2[a2] - (Rm[a2 * 3 + 0] * c1[0] + Rm[a2 * 3 + 1] * c1[1] + Rm[a2 * 3 + 2] * c1[2]);
    out[0]  = Rm[0]; out[1]  = Rm[1]; out[2]  = Rm[2]; out[3]  = tv[0];
    out[4]  = Rm[3]; out[5]  = Rm[4]; out[6]  = Rm[5]; out[7]  = tv[1];
    out[8]  = Rm[6]; out[9]  = Rm[7]; out[10] = Rm[8]; out[11] = tv[2];
    out[12] = 0.0f;  out[13] = 0.0f;  out[14] = 0.0f;  out[15] = 1.0f;
}

// ---------------------------------------------------------------------------
extern "C" void kernel_launch(void* const* d_in, const int* in_sizes, int n_in,
                              void* d_out, int out_size, void* d_ws, size_t ws_size,
                              hipStream_t stream) {
    (void)in_sizes; (void)n_in; (void)out_size; (void)ws_size;
    const float* p1 = (const float*)d_in[0];
    const float* p2 = (const float*)d_in[1];
    float* out = (float*)d_out;
    float* ws  = (float*)d_ws;

    icp_init<<<(NPTS * 3 + 255) / 256, 256, 0, stream>>>(p1, ws);

    for (int it = 0; it < NITERS; ++it) {
        icp_nn<<<NN_BLOCKS, 256, 0, stream>>>(ws + WS_PC, p2, ws + WS_ACC);
        icp_solve<<<1, 1, 0, stream>>>(ws);
        icp_apply<<<(NPTS + 255) / 256, 256, 0, stream>>>(ws);
    }

    icp_final_reduce<<<NPTS / 256, 256, 0, stream>>>(p1, ws);
    icp_final_solve<<<1, 1, 0, stream>>>(ws, out);
}